// Block_87849261072648
// MI455X (gfx1250) — compile-verified
//
#include <hip/hip_runtime.h>
#include <hip/hip_bf16.h>

#define BATCH 8
#define CH    128
#define HH    56
#define WW    56
#define HW    3136      // 56*56
#define RED   32
#define KK    49
#define NG    8
#define GCH   16

typedef float v2f __attribute__((ext_vector_type(2)));
typedef float v8f __attribute__((ext_vector_type(8)));

// ---------------------------------------------------------------------------
// CDNA5 async global->LDS copy (GLOBAL_LOAD_ASYNC_TO_LDS_B32, ASYNCcnt).
// lds_addr: LDS byte address = low 32 bits of a generic pointer to __shared__
// (flat LDS aperture maps addr[31:0] -> LDS offset, ISA 10.2).
// ---------------------------------------------------------------------------
__device__ __forceinline__ void async_g2l_b32(unsigned lds_addr, const void* gptr) {
    asm volatile("global_load_async_to_lds_b32 %0, %1, off"
                 :: "v"(lds_addr), "v"(gptr) : "memory");
}
__device__ __forceinline__ void wait_async0() {
    asm volatile("s_wait_asynccnt 0" ::: "memory");
}

// ---------------------------------------------------------------------------
// Kernel 1: dynamic weight generation.
// wgt[b, 392, h, w] = conv2( relu( bn( conv1(x) ) ) ), all 1x1 convs.
// One block = 32 consecutive pixels (all in same batch since 3136 % 32 == 0).
// ---------------------------------------------------------------------------
__global__ __launch_bounds__(128) void k_weightgen(
    const float* __restrict__ x,
    const float* __restrict__ c1w, const float* __restrict__ c1b,
    const float* __restrict__ bng, const float* __restrict__ bnb,
    const float* __restrict__ bnm, const float* __restrict__ bnv,
    const float* __restrict__ c2w, const float* __restrict__ c2b,
    float* __restrict__ wgt)
{
    __shared__ float xs[CH][32];
    __shared__ float ts[RED][32];
    const int tid = threadIdx.x;
    const int n0  = blockIdx.x * 32;
    const int b   = n0 / HW;
    const int hw  = n0 % HW;
    const float* xb = x + (size_t)b * CH * HW + hw;

    // async-stage the 128x32 input tile straight into LDS
    for (int idx = tid; idx < CH * 32; idx += 128) {
        int c = idx >> 5, n = idx & 31;
        async_g2l_b32((unsigned)(size_t)&xs[c][n], xb + (size_t)c * HW + n);
    }
    wait_async0();
    __syncthreads();

    const int n   = tid & 31;
    const int set = tid >> 5;               // wave-uniform (wave32)

    // conv1 + BN(eval) + ReLU  -> ts[32][32]
    for (int j = 0; j < 8; ++j) {
        int co = set * 8 + j;
        float acc = c1b[co];
        #pragma unroll 8
        for (int k = 0; k < CH; ++k)
            acc = fmaf(c1w[co * CH + k], xs[k][n], acc);
        float sc = bng[co] * rsqrtf(bnv[co] + 1e-5f);
        acc = (acc - bnm[co]) * sc + bnb[co];
        ts[co][n] = fmaxf(acc, 0.0f);
    }
    __syncthreads();

    // conv2 -> wgt (b, 392, h, w)
    float* wb = wgt + (size_t)b * (KK * NG) * HW + hw;
    for (int co = set; co < KK * NG; co += 4) {
        float acc = c2b[co];
        #pragma unroll
        for (int k = 0; k < RED; ++k)
            acc = fmaf(c2w[co * RED + k], ts[k][n], acc);
        wb[(size_t)co * HW + n] = acc;
    }
}

// ---------------------------------------------------------------------------
// Kernel 2: involution gather (7x7, pad 3), per-group dynamic weights.
// Block = (batch, group) x spatial tile of 8 rows x 14 cols, 16 channels.
// Weight tile comes in via the async LDS path; the x halo tile keeps the
// VGPR path because it needs conditional zero padding.  The two regions are
// disjoint, so DS-counter stores and ASYNC-counter loads cannot alias.
// Writes involution output into d_out (kernel 3 then operates in-place).
// ---------------------------------------------------------------------------
#define TH 8
#define TW 14
#define XR (TH + 6)
#define XC (TW + 6)

__global__ __launch_bounds__(256) void k_involution(
    const float* __restrict__ x,
    const float* __restrict__ wgt,
    float* __restrict__ out)
{
    __shared__ float xt[GCH][XR][XC];     // 16*14*20*4 = 17.9 KB
    __shared__ float wt[KK][TH][TW];      // 49*8*14*4  = 21.9 KB
    const int tid = threadIdx.x;
    const int b   = blockIdx.x >> 3;
    const int g   = blockIdx.x & 7;
    const int h0  = blockIdx.y * TH;
    const int w0  = blockIdx.z * TW;

    // dynamic weight tile: async global->LDS
    const float* wb = wgt + ((size_t)b * (KK * NG) + g * KK) * HW;
    for (int idx = tid; idx < KK * TH * TW; idx += 256) {
        int k   = idx / (TH * TW);
        int rem = idx % (TH * TW);
        int r   = rem / TW, cw = rem % TW;
        async_g2l_b32((unsigned)(size_t)&wt[k][r][cw],
                      wb + (size_t)k * HW + (h0 + r) * WW + (w0 + cw));
    }
    // x halo tile (zero padded) via VGPR path
    const float* xb = x + ((size_t)b * CH + g * GCH) * HW;
    for (int idx = tid; idx < GCH * XR * XC; idx += 256) {
        int c   = idx / (XR * XC);
        int rem = idx % (XR * XC);
        int r   = rem / XC, cw = rem % XC;
        int gh = h0 + r - 3, gw = w0 + cw - 3;
        float v = 0.0f;
        if (gh >= 0 && gh < HH && gw >= 0 && gw < WW)
            v = xb[(size_t)c * HW + gh * WW + gw];
        xt[c][r][cw] = v;
    }
    wait_async0();
    __syncthreads();

    const int c   = tid >> 4;   // 0..15  channel within group
    const int l16 = tid & 15;
    float* ob = out + ((size_t)b * CH + g * GCH + c) * HW;
    for (int cw = l16; cw < TW; cw += 16) {
        for (int r = 0; r < TH; ++r) {
            float acc = 0.0f;
            #pragma unroll
            for (int k = 0; k < KK; ++k) {
                int di = k / 7, dj = k % 7;
                acc = fmaf(wt[k][r][cw], xt[c][r + di][cw + dj], acc);
            }
            ob[(h0 + r) * WW + (w0 + cw)] = acc;
        }
    }
}

// ---------------------------------------------------------------------------
// Kernel 3: LayerNorm + MLP (128 -> 512 -> 128, exact-erf GELU) + residual.
// In-place on d_out.  Block = 256 threads (8 waves), 32 pixels (2 WMMA N-tiles).
// GEMMs run on V_WMMA_F32_16X16X4_F32 (fp32 matrix pipe).
// NP=32 halves per-block weight re-read vs NP=16 (W1+W2 = 512 KB streamed per
// block); ~103 KB LDS/block is legal on CDNA5 (320 KB per WGP).
// LDS row stride 40 floats -> half-wave B reads (+2*40 = 80 = 64+16) land on
// banks a+16..a+31 vs a..a+15: conflict-free.
// ---------------------------------------------------------------------------
#define NP   32
#define YS   40
#define DHID 512

__global__ __launch_bounds__(256) void k_ln_mlp(
    float* __restrict__ io,              // involution out -> final out
    const float* __restrict__ x,         // residual
    const float* __restrict__ lnw, const float* __restrict__ lnb,
    const float* __restrict__ w1,  const float* __restrict__ b1,
    const float* __restrict__ w2,  const float* __restrict__ b2)
{
    __shared__ float Y[CH * YS];         // 20.5 KB  (normalized activations)
    __shared__ float Hs[DHID * YS];      // 81.9 KB  (hidden activations)
    __shared__ float ps[8][NP], ps2[8][NP];
    __shared__ float mu_s[NP], rs_s[NP];

    const int tid = threadIdx.x;
    const int n0  = blockIdx.x * NP;
    const int b   = n0 / HW;
    const int hw  = n0 % HW;
    float* iob = io + (size_t)b * CH * HW + hw;

    // ---- async-stage involution output tile into LDS ----
    for (int idx = tid; idx < CH * NP; idx += 256) {
        int c = idx >> 5, n = idx & 31;
        async_g2l_b32((unsigned)(size_t)&Y[c * YS + n], iob + (size_t)c * HW + n);
    }
    wait_async0();
    __syncthreads();

    // ---- LayerNorm over channels ----
    {
        int n = tid & 31, seg = tid >> 5;           // 8 segments of 16 channels
        float s = 0.0f, s2 = 0.0f;
        #pragma unroll
        for (int j = 0; j < 16; ++j) {
            float v = Y[(seg * 16 + j) * YS + n];
            s += v; s2 += v * v;
        }
        ps[seg][n] = s; ps2[seg][n] = s2;
    }
    __syncthreads();
    if (tid < NP) {
        float s = 0.0f, s2 = 0.0f;
        #pragma unroll
        for (int sg = 0; sg < 8; ++sg) { s += ps[sg][tid]; s2 += ps2[sg][tid]; }
        float mu  = s * (1.0f / CH);
        float var = s2 * (1.0f / CH) - mu * mu;
        mu_s[tid] = mu;
        rs_s[tid] = rsqrtf(var + 1e-6f);
    }
    __syncthreads();
    for (int idx = tid; idx < CH * NP; idx += 256) {
        int c = idx >> 5, n = idx & 31;
        Y[c * YS + n] = (Y[c * YS + n] - mu_s[n]) * rs_s[n] * lnw[c] + lnb[c];
    }
    __syncthreads();

    const int lane = tid & 31;
    const int wv   = tid >> 5;           // wave id 0..7 (wave-uniform)
    const int l16  = lane & 15;
    const int lh   = lane >> 4;          // half-wave: selects K pair (ISA A/B layout)
    const int nt   = wv & 1;             // N-tile (0 or 1), wave-uniform
    const int nb   = nt * 16 + l16;      // column in LDS tile

    // ---- GEMM1: Hs = gelu(W1 @ Y + b1), M=512 K=128 N=32 ----
    for (int mt = (wv >> 1); mt < 32; mt += 4) {
        // pull next M-tile's weight rows toward L2/L0 while this tile computes
        if (mt + 4 < 32)
            __builtin_prefetch(w1 + (size_t)((mt + 4) * 16 + l16) * CH, 0, 1);
        v8f acc = {};
        const float* arow = w1 + (size_t)(mt * 16 + l16) * CH + lh * 2;
        #pragma unroll 4
        for (int kb = 0; kb < CH; kb += 4) {
            v2f a;  a.x  = arow[kb];     a.y  = arow[kb + 1];
            v2f bv; bv.x = Y[(kb + lh * 2) * YS + nb];
                    bv.y = Y[(kb + lh * 2 + 1) * YS + nb];
            acc = __builtin_amdgcn_wmma_f32_16x16x4_f32(
                      false, a, false, bv, (short)0, acc, false, false);
        }
        #pragma unroll
        for (int i = 0; i < 8; ++i) {
            int m = mt * 16 + lh * 8 + i;           // D layout: lane half -> row block
            float v = acc[i] + b1[m];
            v = 0.5f * v * (1.0f + erff(v * 0.70710678118654752f));  // exact GELU
            Hs[m * YS + nb] = v;
        }
    }
    __syncthreads();

    // ---- GEMM2: out = W2 @ Hs + b2 + x, M=128 K=512 N=32 ----
    const float* xb = x + (size_t)b * CH * HW + hw;
    for (int mt = (wv >> 1); mt < 8; mt += 4) {
        if (mt + 4 < 8)
            __builtin_prefetch(w2 + (size_t)((mt + 4) * 16 + l16) * DHID, 0, 1);
        v8f acc = {};
        const float* arow = w2 + (size_t)(mt * 16 + l16) * DHID + lh * 2;
        #pragma unroll 4
        for (int kb = 0; kb < DHID; kb += 4) {
            v2f a;  a.x  = arow[kb];     a.y  = arow[kb + 1];
            v2f bv; bv.x = Hs[(kb + lh * 2) * YS + nb];
                    bv.y = Hs[(kb + lh * 2 + 1) * YS + nb];
            acc = __builtin_amdgcn_wmma_f32_16x16x4_f32(
                      false, a, false, bv, (short)0, acc, false, false);
        }
        #pragma unroll
        for (int i = 0; i < 8; ++i) {
            int m = mt * 16 + lh * 8 + i;           // output channel
            float v = acc[i] + b2[m] + xb[(size_t)m * HW + nb];
            iob[(size_t)m * HW + nb] = v;
        }
    }
}

// ---------------------------------------------------------------------------
extern "C" void kernel_launch(void* const* d_in, const int* in_sizes, int n_in,
                              void* d_out, int out_size, void* d_ws, size_t ws_size,
                              hipStream_t stream) {
    (void)in_sizes; (void)n_in; (void)out_size; (void)ws_size;
    const float* x    = (const float*)d_in[0];
    const float* c1w  = (const float*)d_in[1];
    const float* c1b  = (const float*)d_in[2];
    const float* bng  = (const float*)d_in[3];
    const float* bnb  = (const float*)d_in[4];
    const float* bnm  = (const float*)d_in[5];
    const float* bnv  = (const float*)d_in[6];
    const float* c2w  = (const float*)d_in[7];
    const float* c2b  = (const float*)d_in[8];
    const float* lnw  = (const float*)d_in[9];
    const float* lnb  = (const float*)d_in[10];
    const float* w1   = (const float*)d_in[11];
    const float* b1   = (const float*)d_in[12];
    const float* w2   = (const float*)d_in[13];
    const float* b2   = (const float*)d_in[14];
    float* out = (float*)d_out;
    float* wgt = (float*)d_ws;           // 8*392*3136 floats = 39.3 MB

    // 1) dynamic kernel weights
    k_weightgen<<<(BATCH * HW) / 32, 128, 0, stream>>>(
        x, c1w, c1b, bng, bnb, bnm, bnv, c2w, c2b, wgt);
    // 2) involution gather -> d_out
    k_involution<<<dim3(BATCH * NG, HH / TH, WW / TW), 256, 0, stream>>>(
        x, wgt, out);
    // 3) LN + MLP + residual, in-place on d_out (WMMA fp32, NP=32)
    k_ln_mlp<<<(BATCH * HW) / NP, 256, 0, stream>>>(
        out, x, lnw, lnb, w1, b1, w2, b2);
}